// GNNEncoder_47055661695547
// MI455X (gfx1250) — compile-verified
//
#include <hip/hip_runtime.h>

typedef float v2f __attribute__((ext_vector_type(2)));
typedef float v8f __attribute__((ext_vector_type(8)));

#define DIN  128
#define DH   128
#define DOUT 64

// ---------- small utility kernels ----------

__global__ void k_fill(float* __restrict__ p, float v, int n) {
    int i = blockIdx.x * blockDim.x + threadIdx.x;
    if (i < n) p[i] = v;
}

__global__ void k_deg(const int* __restrict__ dst, float* __restrict__ deg, int E) {
    int e = blockIdx.x * blockDim.x + threadIdx.x;
    if (e < E) atomicAdd(&deg[dst[e]], 1.0f);
}

__global__ void k_rsqrt(float* __restrict__ d, int n) {
    int i = blockIdx.x * blockDim.x + threadIdx.x;
    if (i < n) d[i] = rsqrtf(d[i]);   // deg >= 1 always (self-loops)
}

// out[i] = bias[i & dmask]  (row-broadcast bias init; dmask = D-1, D power of 2)
__global__ void k_bias_init(float* __restrict__ out, const float* __restrict__ b,
                            int total, int dmask) {
    int i = blockIdx.x * blockDim.x + threadIdx.x;
    if (i < total) out[i] = b[i & dmask];
}

__global__ void k_relu(float* __restrict__ x, int n) {
    int i = blockIdx.x * blockDim.x + threadIdx.x;
    if (i < n) x[i] = fmaxf(x[i], 0.0f);
}

// ---------- fp32 WMMA GEMM: H[M,N] = X[M,K] @ W[K,N] ----------
// One wave computes one 16x16 C tile via V_WMMA_F32_16X16X4_F32, K stepped by 4.
// A layout (ISA 7.12.2, 32-bit A 16x4): lanes 0-15 hold K=0(v0),K=1(v1) for M=lane;
// lanes 16-31 hold K=2(v0),K=3(v1). B mirrored with N across lanes.
// C/D layout: VGPR r -> row (r + 8*half), col = lane&15.
__global__ void k_gemm_wmma(const float* __restrict__ X, const float* __restrict__ W,
                            float* __restrict__ H, int M, int K, int N, int tiles) {
    int wave = (blockIdx.x * blockDim.x + threadIdx.x) >> 5;
    if (wave >= tiles) return;                 // wave-uniform: EXEC stays all-ones
    int lane   = threadIdx.x & 31;
    int ntiles = N >> 4;
    int m0 = (wave / ntiles) << 4;
    int n0 = (wave % ntiles) << 4;
    int half = lane >> 4;                      // 0: lanes 0-15, 1: lanes 16-31
    int l    = lane & 15;

    v8f c = {0.f, 0.f, 0.f, 0.f, 0.f, 0.f, 0.f, 0.f};

    const float* arow = X + (size_t)(m0 + l) * K + 2 * half;  // pairs (K, K+1)
    const float* bcol = W + (size_t)(2 * half) * N + n0 + l;

    for (int s = 0; s < K; s += 4) {
        v2f a = *(const v2f*)(arow + s);                      // 8B aligned (s,2*half even)
        v2f b = { bcol[(size_t)s * N], bcol[(size_t)(s + 1) * N] };
        c = __builtin_amdgcn_wmma_f32_16x16x4_f32(
                /*neg_a=*/false, a, /*neg_b=*/false, b,
                /*c_mod=*/(short)0, c, /*reuse_a=*/false, /*reuse_b=*/false);
    }

    float* hp = H + (size_t)(m0 + 8 * half) * N + n0 + l;
#pragma unroll
    for (int r = 0; r < 8; ++r) hp[(size_t)r * N] = c[r];
}

// ---------- edge gather * norm -> atomic scatter-add ----------
// One wave per edge (incl. implicit self-loops at wave >= E). Lanes cover the
// feature dim: D/32 floats per lane, coalesced vector gather from H (L2-resident
// on MI455X: H fits in the 192MB L2), then global_atomic_add_f32 per element.
template <int D, int PL>
__global__ void k_scatter(const float* __restrict__ H, const int* __restrict__ src,
                          const int* __restrict__ dst, const float* __restrict__ dinv,
                          float* __restrict__ out, int E, int Nn) {
    int wave = (blockIdx.x * blockDim.x + threadIdx.x) >> 5;
    if (wave >= E + Nn) return;
    int lane = threadIdx.x & 31;
    int s, d;
    if (wave < E) { s = src[wave]; d = dst[wave]; }
    else          { s = d = wave - E; }                 // self-loop
    float nrm = dinv[s] * dinv[d];

    const float* hp = H + (size_t)s * D + lane * PL;
    float*       op = out + (size_t)d * D + lane * PL;

    float v[PL];
    if constexpr (PL == 4) {
        float4 t = *(const float4*)hp;
        v[0] = t.x; v[1] = t.y; v[2] = t.z; v[3] = t.w;
    } else {
        float2 t = *(const float2*)hp;
        v[0] = t.x; v[1] = t.y;
    }
#pragma unroll
    for (int j = 0; j < PL; ++j) atomicAdd(&op[j], v[j] * nrm);
}

// ---------- launcher ----------

extern "C" void kernel_launch(void* const* d_in, const int* in_sizes, int n_in,
                              void* d_out, int out_size, void* d_ws, size_t ws_size,
                              hipStream_t stream) {
    const float* X    = (const float*)d_in[0];   // [N,128]
    const int*   eidx = (const int*)  d_in[1];   // [2,E]
    const float* W1   = (const float*)d_in[2];   // [128,128]
    const float* b1   = (const float*)d_in[3];   // [128]
    const float* W2   = (const float*)d_in[4];   // [128,64]
    const float* b2   = (const float*)d_in[5];   // [64]
    float* out = (float*)d_out;                  // [N,64]

    const int N = in_sizes[0] / DIN;             // 50000
    const int E = in_sizes[1] / 2;               // 800000
    const int* esrc = eidx;
    const int* edst = eidx + E;

    // workspace: dinv | H (N x 128) | A (N x 128 accumulator / activations)
    char* ws = (char*)d_ws;
    float* dinv = (float*)ws;
    size_t off  = (((size_t)N * 4) + 255) & ~(size_t)255;
    float* H = (float*)(ws + off);
    float* A = (float*)(ws + off + (size_t)N * DH * 4);

    const int BT = 256;
    auto cdiv = [](long a, long b) { return (int)((a + b - 1) / b); };

    // 1-3: dinv[i] = rsqrt(1 + #incoming-edges)
    k_fill <<<cdiv(N, BT), BT, 0, stream>>>(dinv, 1.0f, N);
    k_deg  <<<cdiv(E, BT), BT, 0, stream>>>(edst, dinv, E);
    k_rsqrt<<<cdiv(N, BT), BT, 0, stream>>>(dinv, N);

    // 4: A = broadcast(b1)  (bias folded into the scatter accumulator)
    k_bias_init<<<cdiv((long)N * DH, BT), BT, 0, stream>>>(A, b1, N * DH, DH - 1);

    // 5: H = X @ W1  (fp32 WMMA)
    int tiles1 = (N / 16) * (DH / 16);
    k_gemm_wmma<<<cdiv((long)tiles1 * 32, BT), BT, 0, stream>>>(X, W1, H, N, DIN, DH, tiles1);

    // 6: A += norm * H[src]  scattered to dst (+ self-loops)
    long waves = (long)E + N;
    k_scatter<DH, DH / 32><<<cdiv(waves * 32, BT), BT, 0, stream>>>(H, esrc, edst, dinv, A, E, N);

    // 7: A = relu(A)
    k_relu<<<cdiv((long)N * DH, BT), BT, 0, stream>>>(A, N * DH);

    // 8: H[:, :64] = A @ W2  (fp32 WMMA)
    int tiles2 = (N / 16) * (DOUT / 16);
    k_gemm_wmma<<<cdiv((long)tiles2 * 32, BT), BT, 0, stream>>>(A, W2, H, N, DH, DOUT, tiles2);

    // 9: out = broadcast(b2)
    k_bias_init<<<cdiv((long)N * DOUT, BT), BT, 0, stream>>>(out, b2, N * DOUT, DOUT - 1);

    // 10: out += norm * H[src] scattered to dst (+ self-loops)
    k_scatter<DOUT, DOUT / 32><<<cdiv(waves * 32, BT), BT, 0, stream>>>(H, esrc, edst, dinv, out, E, N);
}